// GenericShift3D_4904852652808
// MI455X (gfx1250) — compile-verified
//
#include <hip/hip_runtime.h>

// GenericShift3D: out[b,c,z,y,x] = in[b,c,z+oz,y+oy,x+ox] (zero outside volume)
// (oz,oy,ox) from channel group i = c/4:  oz=i/9-1, oy=(i/3)%3-1, ox=i%3-1
// Shapes: B=4, C=108, D=16, H=128, W=128, fp32. cps=4, rem=0.
//
// Pure streaming op: 0.906 GB traffic -> ~39 us floor at 23.3 TB/s (MI455X).
// Key structure: blockIdx.y = bc = b*108+c, so ALL channel/shift math
// (c = bc%108, i, oz/oy/ox, the ox==0-vs-misaligned branch, and the sz bound)
// is wave-uniform -> SALU, once per wave. Per-thread VALU is just shifts/masks.
// b128 non-temporal load/store (working set 906MB >> 192MB L2, single touch).

typedef float v4f __attribute__((ext_vector_type(4)));

#define SC 108
#define SD 16
#define SH 128
#define SW 128
#define W4 (SW / 4)           // 32 float4 per row
#define SLICE4 (SD * SH * W4) // 65536 float4 per (b,c) slice

__global__ __launch_bounds__(256) void shift3d_f32_kernel(
    const float* __restrict__ in, float* __restrict__ out)
{
    const int bc  = blockIdx.y;                       // b*108 + c  (uniform)
    const int tid = blockIdx.x * 256 + threadIdx.x;   // 0..65535 within slice

    const int w4 = tid & (W4 - 1);        // 5 bits, per-thread
    const int y  = (tid >> 5) & (SH - 1); // 7 bits, bits 5-7 per-thread
    const int z  = tid >> 12;             // uniform per block (block spans bits 0-7)

    // ---- all scalar (block-uniform) ----
    const int c  = bc % SC;
    const int i  = c >> 2;                // 0..26
    const int oz = (i / 9) - 1;
    const int t9 = i % 9;
    const int oy = (t9 / 3) - 1;
    const int ox = (t9 % 3) - 1;
    const int sz = z + oz;                // uniform
    // ------------------------------------

    const int sy = y + oy;

    v4f r = (v4f){0.f, 0.f, 0.f, 0.f};

    if ((unsigned)sz < (unsigned)SD && (unsigned)sy < (unsigned)SH) {
        const float* __restrict__ src =
            in + ((((long)bc * SD + sz) * SH + sy) * SW);
        const int x0 = (w4 << 2) + ox;

        if (ox == 0) {
            // 16B-aligned source: single b128 non-temporal load
            r = __builtin_nontemporal_load(
                    reinterpret_cast<const v4f*>(src + x0));
        } else {
            // shifted by one element: 4 coalesced dword loads, clamp row edges.
            // Only x0 (== -1, ox=-1,w4=0) or x0+3 (== 128, ox=+1,w4=31) can be OOB.
            r.x = ((unsigned)(x0 + 0) < (unsigned)SW) ? src[x0 + 0] : 0.f;
            r.y = src[x0 + 1];
            r.z = src[x0 + 2];
            r.w = ((unsigned)(x0 + 3) < (unsigned)SW) ? src[x0 + 3] : 0.f;
        }
    }

    // aligned b128 non-temporal store; grid exactly covers the output
    const long oidx = ((long)bc << 16) + tid;   // bc*SLICE4 + tid
    __builtin_nontemporal_store(r, reinterpret_cast<v4f*>(out) + oidx);
}

extern "C" void kernel_launch(void* const* d_in, const int* in_sizes, int n_in,
                              void* d_out, int out_size, void* d_ws, size_t ws_size,
                              hipStream_t stream)
{
    const float* x = (const float*)d_in[0];
    float* out = (float*)d_out;

    // 432 (b,c) slices, each 65536 float4 = 256 blocks of 256 threads
    dim3 grid(SLICE4 / 256, 4 * SC, 1);
    shift3d_f32_kernel<<<grid, dim3(256, 1, 1), 0, stream>>>(x, out);
}